// CoreGCN_53085795778683
// MI455X (gfx1250) — compile-verified
//
#include <hip/hip_runtime.h>
#include <hip/hip_bf16.h>

typedef __attribute__((ext_vector_type(2))) float v2f;
typedef __attribute__((ext_vector_type(8))) float v8f;

#define FDIM 128

// ---------------- degree / norm ----------------

__global__ void deg_init_kernel(float* __restrict__ deg, int n) {
    int i = blockIdx.x * blockDim.x + threadIdx.x;
    if (i < n) deg[i] = 1.0f;   // self-loop weight
}

__global__ void deg_accum_kernel(const int* __restrict__ col,
                                 const float* __restrict__ ew,
                                 float* __restrict__ deg, int e) {
    int i = blockIdx.x * blockDim.x + threadIdx.x;
    if (i < e) unsafeAtomicAdd(&deg[col[i]], ew[i]);
}

__global__ void deg_to_dis_kernel(float* __restrict__ deg, int n) {
    int i = blockIdx.x * blockDim.x + threadIdx.x;
    if (i < n) {
        float d = deg[i];
        deg[i] = (d > 0.0f) ? rsqrtf(fmaxf(d, 1e-30f)) : 0.0f;
    }
}

__global__ void edge_norm_kernel(const int* __restrict__ row,
                                 const int* __restrict__ col,
                                 const float* __restrict__ ew,
                                 const float* __restrict__ dis,
                                 float* __restrict__ nrm, int e) {
    int i = blockIdx.x * blockDim.x + threadIdx.x;
    if (i < e) nrm[i] = dis[row[i]] * ew[i] * dis[col[i]];
}

// ---------------- WMMA GEMM: C[N,128] = X[N,128] @ W[128,128] ----------------
// One block (256 thr = 8 waves) per 16-row tile of X. Wave w owns output
// columns [16w, 16w+16). K-loop: 32 x v_wmma_f32_16x16x4_f32.
// Fragment layouts per CDNA5 ISA 7.12.2 (f32, wave32):
//   A 16x4 : v0 = A[m][k+2*kh], v1 = A[m][k+2*kh+1]   (m = lane&15, kh = lane>>4)
//   B 4x16 : v0 = B[k+2*kh][n], v1 = B[k+2*kh+1][n]   (n = lane&15)
//   C 16x16: vr -> C[8*kh + r][n]

__global__ void __launch_bounds__(256)
gemm_wmma_kernel(const float* __restrict__ X, const float* __restrict__ W,
                 float* __restrict__ C) {
    __shared__ float sW[FDIM * FDIM];   // 64 KB
    __shared__ float sX[16 * FDIM];     //  8 KB
    const int t = threadIdx.x;

    // Stage W (whole 128x128) and the 16x128 X tile in LDS, float4 moves.
    const float4* Wv = (const float4*)W;
    float4* sWv = (float4*)sW;
    for (int i = t; i < FDIM * FDIM / 4; i += 256) sWv[i] = Wv[i];
    const float4* Xv = (const float4*)(X + (size_t)blockIdx.x * 16 * FDIM);
    float4* sXv = (float4*)sX;
    for (int i = t; i < 16 * FDIM / 4; i += 256) sXv[i] = Xv[i];
    __syncthreads();

    const int wave = t >> 5;         // 0..7 -> column tile
    const int lane = t & 31;
    const int m    = lane & 15;      // row within tile / col within tile
    const int kh   = lane >> 4;      // K-half select
    const int n0   = wave * 16;

    v8f acc = {};
    #pragma unroll
    for (int k = 0; k < FDIM; k += 4) {
        v2f a, b;
        a.x = sX[m * FDIM + k + 2 * kh];
        a.y = sX[m * FDIM + k + 2 * kh + 1];
        b.x = sW[(k + 2 * kh) * FDIM + n0 + m];
        b.y = sW[(k + 2 * kh + 1) * FDIM + n0 + m];
        acc = __builtin_amdgcn_wmma_f32_16x16x4_f32(
                  false, a, false, b, (short)0, acc, false, false);
    }

    const size_t row0 = (size_t)blockIdx.x * 16;
    #pragma unroll
    for (int r = 0; r < 8; ++r) {
        C[(row0 + 8 * kh + r) * FDIM + n0 + m] = acc[r];
    }
}

// ---------------- aggregation ----------------

// out[i,f] = XW[i,f] * dis[i]^2 + bias[f]   (self-loop term + bias)
__global__ void agg_init_kernel(const float* __restrict__ XW,
                                const float* __restrict__ dis,
                                const float* __restrict__ bias,
                                float* __restrict__ out, int n) {
    long long i = (long long)blockIdx.x * blockDim.x + threadIdx.x;
    long long total = (long long)n * FDIM;
    if (i >= total) return;
    int node = (int)(i >> 7);
    int f = (int)(i & (FDIM - 1));
    float d = dis[node];
    out[i] = XW[i] * d * d + bias[f];
}

// 32 threads per edge, each thread scatters 4 features (float4 gather + 4 fp32 atomics)
__global__ void scatter_kernel(const int* __restrict__ row,
                               const int* __restrict__ col,
                               const float* __restrict__ nrm,
                               const float* __restrict__ H,
                               float* __restrict__ out, int e) {
    long long idx = (long long)blockIdx.x * blockDim.x + threadIdx.x;
    long long total = (long long)e * 32;
    if (idx >= total) return;
    int ei = (int)(idx >> 5);
    int f4 = (int)(idx & 31) * 4;
    int r = row[ei];
    int c = col[ei];
    float w = nrm[ei];
    float4 h = *(const float4*)(H + (size_t)r * FDIM + f4);
    float* dst = out + (size_t)c * FDIM + f4;
    unsafeAtomicAdd(dst + 0, h.x * w);
    unsafeAtomicAdd(dst + 1, h.y * w);
    unsafeAtomicAdd(dst + 2, h.z * w);
    unsafeAtomicAdd(dst + 3, h.w * w);
}

// ---------------- batchnorm ----------------

__global__ void zero_stats_kernel(float* __restrict__ s) {
    s[threadIdx.x] = 0.0f;   // launch with 256 threads: sum[128] + sumsq[128]
}

#define BN_CHUNK 256
__global__ void bn_stats_kernel(const float* __restrict__ H,
                                float* __restrict__ stats, int n) {
    int f = threadIdx.x;            // 128 threads = 128 features
    int r0 = blockIdx.x * BN_CHUNK;
    int r1 = r0 + BN_CHUNK; if (r1 > n) r1 = n;
    float s = 0.0f, s2 = 0.0f;
    for (int r = r0; r < r1; ++r) {
        float v = H[(size_t)r * FDIM + f];
        s += v;
        s2 += v * v;
    }
    unsafeAtomicAdd(&stats[f], s);
    unsafeAtomicAdd(&stats[FDIM + f], s2);
}

__global__ void bn_finalize_kernel(const float* __restrict__ stats,
                                   const float* __restrict__ gamma,
                                   const float* __restrict__ beta,
                                   float* __restrict__ ss, float invN) {
    int f = threadIdx.x;  // 128
    float mean = stats[f] * invN;
    float var = stats[FDIM + f] * invN - mean * mean;
    float sc = gamma[f] * rsqrtf(var + 1e-5f);
    ss[f] = sc;
    ss[FDIM + f] = beta[f] - mean * sc;
}

__global__ void bn_apply_relu_kernel(const float* __restrict__ in,
                                     const float* __restrict__ ss,
                                     float* __restrict__ out, long long total) {
    long long i = (long long)blockIdx.x * blockDim.x + threadIdx.x;
    if (i >= total) return;
    int f = (int)(i & (FDIM - 1));
    float v = in[i] * ss[f] + ss[FDIM + f];
    out[i] = v > 0.0f ? v : 0.0f;
}

// ---------------- launch ----------------

static inline int ceil_div_i(long long a, int b) { return (int)((a + b - 1) / b); }

extern "C" void kernel_launch(void* const* d_in, const int* in_sizes, int n_in,
                              void* d_out, int out_size, void* d_ws, size_t ws_size,
                              hipStream_t stream) {
    const float* x      = (const float*)d_in[0];
    const int*   ei     = (const int*)d_in[1];     // [2,E]: row = ei[0:E), col = ei[E:2E)
    const float* ew     = (const float*)d_in[2];
    const float* W1     = (const float*)d_in[3];
    const float* b1     = (const float*)d_in[4];
    const float* gamma1 = (const float*)d_in[5];
    const float* beta1  = (const float*)d_in[6];
    const float* W2     = (const float*)d_in[7];
    const float* b2     = (const float*)d_in[8];
    const float* gamma2 = (const float*)d_in[9];
    const float* beta2  = (const float*)d_in[10];

    const int N = in_sizes[0] / FDIM;
    const int E = in_sizes[2];
    const int* erow = ei;
    const int* ecol = ei + E;

    // workspace carve-up
    float* p    = (float*)d_ws;
    float* dis  = p;                 p += N;
    float* nrm  = p;                 p += E;
    float* bufA = p;                 p += (size_t)N * FDIM;
    float* bufB = p;                 p += (size_t)N * FDIM;
    float* stats = p;                p += 2 * FDIM;
    float* bnp   = p;                p += 2 * FDIM;
    float* out  = (float*)d_out;

    const long long totNF = (long long)N * FDIM;
    const long long totE32 = (long long)E * 32;
    const float invN = 1.0f / (float)N;

    // --- gcn_norm ---
    deg_init_kernel<<<ceil_div_i(N, 256), 256, 0, stream>>>(dis, N);
    deg_accum_kernel<<<ceil_div_i(E, 256), 256, 0, stream>>>(ecol, ew, dis, E);
    deg_to_dis_kernel<<<ceil_div_i(N, 256), 256, 0, stream>>>(dis, N);
    edge_norm_kernel<<<ceil_div_i(E, 256), 256, 0, stream>>>(erow, ecol, ew, dis, nrm, E);

    // --- layer 1 ---
    gemm_wmma_kernel<<<N / 16, 256, 0, stream>>>(x, W1, bufA);           // XW1 -> A
    agg_init_kernel<<<ceil_div_i(totNF, 256), 256, 0, stream>>>(bufA, dis, b1, bufB, N);
    scatter_kernel<<<ceil_div_i(totE32, 256), 256, 0, stream>>>(erow, ecol, nrm, bufA, bufB, E);
    zero_stats_kernel<<<1, 256, 0, stream>>>(stats);
    bn_stats_kernel<<<ceil_div_i(N, BN_CHUNK), FDIM, 0, stream>>>(bufB, stats, N);
    bn_finalize_kernel<<<1, FDIM, 0, stream>>>(stats, gamma1, beta1, bnp, invN);
    bn_apply_relu_kernel<<<ceil_div_i(totNF, 256), 256, 0, stream>>>(bufB, bnp, bufA, totNF); // h1 -> A

    // --- layer 2 ---
    gemm_wmma_kernel<<<N / 16, 256, 0, stream>>>(bufA, W2, bufB);        // XW2 -> B
    agg_init_kernel<<<ceil_div_i(totNF, 256), 256, 0, stream>>>(bufB, dis, b2, out, N);
    scatter_kernel<<<ceil_div_i(totE32, 256), 256, 0, stream>>>(erow, ecol, nrm, bufB, out, E);
    zero_stats_kernel<<<1, 256, 0, stream>>>(stats);
    bn_stats_kernel<<<ceil_div_i(N, BN_CHUNK), FDIM, 0, stream>>>(out, stats, N);
    bn_finalize_kernel<<<1, FDIM, 0, stream>>>(stats, gamma2, beta2, bnp, invN);
    bn_apply_relu_kernel<<<ceil_div_i(totNF, 256), 256, 0, stream>>>(out, bnp, out, totNF);
}